// EdgeGuidedFusionLoss_23313082482722
// MI455X (gfx1250) — compile-verified
//
#include <hip/hip_runtime.h>
#include <math.h>

// ---------------------------------------------------------------------------
// EdgeGuidedFusionLoss forward for MI455X (gfx1250, wave32).
//
// Dominant work: debiased Sinkhorn divergence on 4096-pt 3D clouds (B=2).
// Cost tiles are recomputed on the fly with V_WMMA_F32_16X16X4_F32
// (16x16 tile of dot products per instruction, K=4 = xyz + zero pad), so the
// three 4096x4096 cost matrices never touch HBM. Log-sum-exp is done online
// per lane (8 rows/lane per the 16x16 f32 D layout), halves merged with a
// lane-16 shuffle. Chunk staging into LDS is double-buffered through the
// Tensor Data Mover (tensor_load_to_lds + s_wait_tensorcnt) so the DMA of the
// next 24 KB overlaps the exp-bound compute on the current chunk.
//
// Workspace requirement: ~2.3 MB (see offsets below).
// ---------------------------------------------------------------------------

typedef __attribute__((ext_vector_type(2))) float v2f;
typedef __attribute__((ext_vector_type(8))) float v8f;
typedef __attribute__((ext_vector_type(4))) unsigned int v4u;
typedef __attribute__((ext_vector_type(4))) int v4i;
typedef __attribute__((ext_vector_type(8))) int v8i;

#define NPTS   4096
#define CHUNK  1024
#define NCHUNK (NPTS / CHUNK)
#define HW_    65536
#define EPS_R  0.0025f        // blur**p = 0.05**2
#define INV_EPS 400.0f
#define LOGN   8.3177661667f  // ln(4096)

// TDM availability (device pass only; host pass takes the fallback branch)
#if defined(__HIP_DEVICE_COMPILE__) && defined(__has_builtin)
#if __has_builtin(__builtin_amdgcn_tensor_load_to_lds)
#define HAVE_TDM 1
#endif
#endif
#ifndef HAVE_TDM
#define HAVE_TDM 0
#endif

// ---- workspace layout (bytes) ----
static constexpr size_t OFF_YPTS  = 0;                    // 2*4096 float4 = 131072
static constexpr size_t OFF_PPTS  = 131072;               // 2*4096 float4
static constexpr size_t OFF_POT   = 262144;               // 6 arrays * 2*4096 float2 = 393216
static constexpr size_t OFF_ACC   = OFF_POT + 393216;     // 16 floats
static constexpr size_t OFF_VALID = OFF_ACC + 64;         // 2 ints
static constexpr size_t OFF_TMP   = OFF_VALID + 128;      // blur temp: 393216 floats
// total = OFF_TMP + 1572864 ~= 2.3 MB

// ---------------------------------------------------------------------------
#if HAVE_TDM
// Issue a 1-D TDM copy: nElem8 x 8-byte elements, global -> LDS.
// D# built per CDNA5 ISA 8.3/8.4 (group0: count/lds/global/type=2;
// group1: data_size=8B, tensor_dim0 = tile_dim0 = nElem8, dim1 = 1).
__device__ __forceinline__ void tdm_load_1d(unsigned lds_off, const void* gptr,
                                            unsigned nElem8) {
  unsigned long long ga = (unsigned long long)(size_t)gptr;
  v4u g0;
  g0[0] = 1u;                                            // count=1 (user D#)
  g0[1] = lds_off;                                       // lds_addr
  g0[2] = (unsigned)(ga & 0xFFFFFFFFu);                  // global_addr[31:0]
  g0[3] = (unsigned)((ga >> 32) & 0x01FFFFFFu) | 0x80000000u;  // [56:32] | type=2
  v8i g1;
  g1[0] = (int)(3u << 16);                               // data_size = 8 bytes
  g1[1] = (int)((nElem8 & 0xFFFFu) << 16);               // tensor_dim0[15:0]
  g1[2] = (int)((nElem8 >> 16) | (1u << 16));            // tensor_dim0[31:16] | tensor_dim1=1
  g1[3] = (int)((nElem8 & 0xFFFFu) << 16);               // tile_dim0
  g1[4] = 0;                                             // tile_dim1/2 unused
  g1[5] = (int)nElem8;                                   // tensor_dim0_stride[31:0]
  g1[6] = 0;
  g1[7] = 0;
  v4i z4 = {0, 0, 0, 0};
#if __clang_major__ >= 23
  v8i z8 = {0, 0, 0, 0, 0, 0, 0, 0};
  __builtin_amdgcn_tensor_load_to_lds(g0, g1, z4, z4, z8, 0);
#else
  __builtin_amdgcn_tensor_load_to_lds(g0, g1, z4, z4, 0);
#endif
}
#endif

// ---------------------------------------------------------------------------
__global__ __launch_bounds__(256) void zero_ws_k(float* __restrict__ p, int n) {
  int i = blockIdx.x * 256 + threadIdx.x;
  if (i < n) p[i] = 0.0f;
}

// count mask > 0.1 per sample (blocks never straddle samples: 65536 % 256 == 0)
__global__ __launch_bounds__(256) void count_valid_k(const float* __restrict__ mask,
                                                     int* __restrict__ valid) {
  int i = blockIdx.x * 256 + threadIdx.x;       // < 2*65536
  int b = i >> 16;
  bool p = mask[i] > 0.1f;
  unsigned long long bal = __ballot(p);
  if ((threadIdx.x & 31) == 0)
    atomicAdd(&valid[b], (int)__popcll(bal));
}

// gather 4096 points/sample (pet_s, y_s = (y*mask)_s) with 0.5||.||^2 in .w,
// and seed the six {0.5||.||^2, pot=0} float2 potential buffers.
__global__ __launch_bounds__(256) void gather_pts_k(const float* __restrict__ y,
                                                    const float* __restrict__ pet,
                                                    const float* __restrict__ mask,
                                                    const int* __restrict__ idx,
                                                    float4* __restrict__ yPts,
                                                    float4* __restrict__ pPts,
                                                    float2* __restrict__ pot) {
  int k = blockIdx.x * 256 + threadIdx.x;       // < 2*4096
  int b  = k >> 12;
  int id = idx[k];
  float m  = mask[b * HW_ + id];
  float px = pet[(b * 3 + 0) * HW_ + id];
  float py = pet[(b * 3 + 1) * HW_ + id];
  float pz = pet[(b * 3 + 2) * HW_ + id];
  float yx = y[(b * 3 + 0) * HW_ + id] * m;
  float yy = y[(b * 3 + 1) * HW_ + id] * m;
  float yz = y[(b * 3 + 2) * HW_ + id] * m;
  float sap = 0.5f * (px * px + py * py + pz * pz);
  float say = 0.5f * (yx * yx + yy * yy + yz * yz);
  pPts[k] = make_float4(px, py, pz, sap);
  yPts[k] = make_float4(yx, yy, yz, say);
  float2 sy = make_float2(say, 0.0f);
  float2 sp = make_float2(sap, 0.0f);
  // arrays: 0=f(y) 1=g(pet) 2=px0(y) 3=px1(y) 4=py0(pet) 5=py1(pet)
  pot[0 * 8192 + k] = sy;
  pot[2 * 8192 + k] = sy;
  pot[3 * 8192 + k] = sy;
  pot[1 * 8192 + k] = sp;
  pot[4 * 8192 + k] = sp;
  pot[5 * 8192 + k] = sp;
}

// ---------------------------------------------------------------------------
// One Sinkhorn softmin pass:
//   T_j = -eps * ( LSE_i( (pot_i - C_ij)/eps ) - log n ),
//   C_ij = max(0.5||A_i||^2 + 0.5||B_j||^2 - A_i.B_j, 0)
// Dot tiles via V_WMMA_F32_16X16X4_F32; LDS double-buffered via TDM.
// potOut ? potOut_j.y = mix ? 0.5*(potPrev_j.y + T_j) : T_j
// acc    ? atomicAdd(acc, gate * accWeight * T_j / n)
// gridDim = (32, B); 8 waves/block, one 16-column tile per wave.
// ---------------------------------------------------------------------------
__global__ __launch_bounds__(256) void sink_pass_k(
    const float4* __restrict__ ptsAg, const float4* __restrict__ ptsBg,
    const float2* __restrict__ potSrcg, const float2* __restrict__ potPrevg,
    float2* __restrict__ potOutg, float* __restrict__ accg,
    const int* __restrict__ validg, float accWeight, int mix) {
  __shared__ float4 sPos[2][CHUNK];   // 2 x 16 KB  (x,y,z,sa)
  __shared__ float2 sSP[2][CHUNK];    // 2 x  8 KB  (sa, pot)

  const int b = blockIdx.y;
  const float4* ptsA   = ptsAg + (size_t)b * NPTS;
  const float4* ptsB   = ptsBg + (size_t)b * NPTS;
  const float2* potSrc = potSrcg + (size_t)b * NPTS;

  const int tid  = threadIdx.x;
  const int lane = tid & 31;
  const int wave = tid >> 5;
  const int half = lane >> 4;                                   // 0: K0/K1, 1: K2/K3
  const int nIdx = (blockIdx.x * 8 + wave) * 16 + (lane & 15);  // output column j

  // B fragment (4x16 K-major: v0 = K0|K2 row, v1 = K1|K3 row)
  float4 pb = ptsB[nIdx];
  v2f bFrag;
  bFrag[0] = half ? pb.z : pb.x;
  bFrag[1] = half ? 0.0f : pb.y;
  const float sb = pb.w;

  float runM = -3.40282347e38f;
  float runS = 0.0f;

#if HAVE_TDM
  if (wave == 0) {   // prime the pipeline: DMA chunk 0 into buffer 0
    tdm_load_1d((unsigned)(size_t)&sPos[0][0], ptsA, CHUNK * 2);
    tdm_load_1d((unsigned)(size_t)&sSP[0][0], potSrc, CHUNK);
  }
#endif
  int cur = 0;

  for (int chunk = 0; chunk < NCHUNK; ++chunk) {
#if HAVE_TDM
    if (wave == 0) __builtin_amdgcn_s_wait_tensorcnt(0);
    __syncthreads();                       // buffer `cur` published to all waves
    if (wave == 0 && chunk + 1 < NCHUNK) { // overlap: DMA next chunk into cur^1
      tdm_load_1d((unsigned)(size_t)&sPos[cur ^ 1][0],
                  ptsA + (chunk + 1) * CHUNK, CHUNK * 2);
      tdm_load_1d((unsigned)(size_t)&sSP[cur ^ 1][0],
                  potSrc + (chunk + 1) * CHUNK, CHUNK);
    }
#else
    for (int t = tid; t < CHUNK; t += 256) {
      sPos[cur][t] = ptsA[chunk * CHUNK + t];
      sSP[cur][t]  = potSrc[chunk * CHUNK + t];
    }
    if (chunk + 1 < NCHUNK)
      __builtin_prefetch((const void*)(ptsA + (chunk + 1) * CHUNK + tid * 4), 0, 0);
    __syncthreads();
#endif

    for (int tb = 0; tb < CHUNK; tb += 16) {
      // A fragment (16x4: lanes 0-15 -> K0,K1 of row M=lane; 16-31 -> K2,K3)
      float4 pa = sPos[cur][tb + (lane & 15)];
      v2f aFrag;
      aFrag[0] = half ? pa.z : pa.x;
      aFrag[1] = half ? 0.0f : pa.y;

      v8f czero = {0.f, 0.f, 0.f, 0.f, 0.f, 0.f, 0.f, 0.f};
      v8f d = __builtin_amdgcn_wmma_f32_16x16x4_f32(
          false, aFrag, false, bFrag, (short)0, czero, false, false);

      // D layout: vgpr r holds (M = r + 8*half, N = lane&15)
      float arg[8];
      float mt = -3.40282347e38f;
#pragma unroll
      for (int r = 0; r < 8; ++r) {
        int iL = tb + r + half * 8;
        float2 sp = sSP[cur][iL];                      // {sa_i, pot_i} contiguous
        float C = fmaxf(sp.x + sb - d[r], 0.0f);
        arg[r] = (sp.y - C) * INV_EPS;
        mt = fmaxf(mt, arg[r]);
      }
      float nm = fmaxf(runM, mt);
      float s = 0.0f;
#pragma unroll
      for (int r = 0; r < 8; ++r) s += __expf(arg[r] - nm);
      runS = runS * __expf(runM - nm) + s;
      runM = nm;
    }
    cur ^= 1;
  }

  // merge the two half-columns (rows 0-7 vs 8-15)
  float oM = __shfl_xor(runM, 16, 32);
  float oS = __shfl_xor(runS, 16, 32);
  float M2 = fmaxf(runM, oM);
  float S2 = runS * __expf(runM - M2) + oS * __expf(oM - M2);
  float T  = -EPS_R * (M2 + __logf(S2) - LOGN);

  if (lane < 16) {
    if (potOutg) {
      float v = T;
      if (mix) v = 0.5f * (potPrevg[(size_t)b * NPTS + nIdx].y + T);
      potOutg[(size_t)b * NPTS + nIdx].y = v;
    }
    if (accg) {
      float w = accWeight;
      if (validg && validg[b] < 32) w = 0.0f;
      atomicAdd(accg, w * T * (1.0f / NPTS));
    }
  }
}

// ---------------------------------------------------------------------------
// beta*L_out + L_intensity over B*3*H*W
__global__ __launch_bounds__(256) void elem_loss_k(const float* __restrict__ y,
                                                   const float* __restrict__ pet,
                                                   const float* __restrict__ mask,
                                                   float* __restrict__ acc) {
  int i = blockIdx.x * 256 + threadIdx.x;       // < 393216
  int plane = i >> 16;
  int b = plane / 3;
  int pix = i & 65535;
  float m  = mask[b * HW_ + pix];
  float yv = y[i], pv = pet[i];
  const float invN = 1.0f / 393216.0f;
  float a0 = yv * (1.0f - m);
  float a1 = m * (yv - pv);
  float v = 0.3f * invN * a0 * a0 + 2.5f * invN * a1 * a1;
  for (int o = 16; o > 0; o >>= 1) v += __shfl_down(v, o, 32);
  if ((threadIdx.x & 31) == 0) atomicAdd(acc, v);
}

__device__ __constant__ float GW[9] = {
    1.3383062e-4f, 4.4318615e-3f, 5.3990968e-2f, 2.4197073e-1f,
    3.9894229e-1f, 2.4197073e-1f, 5.3990968e-2f, 4.4318615e-3f, 1.3383062e-4f};

__device__ __forceinline__ int refl256(int p) {
  p = (p < 0) ? (-p - 1) : p;
  return (p > 255) ? (511 - p) : p;
}

__global__ __launch_bounds__(256) void blur_v_k(const float* __restrict__ y,
                                                float* __restrict__ tmp) {
  int i = blockIdx.x * 256 + threadIdx.x;       // < 393216
  int plane = i >> 16;
  int p = i & 65535;
  int h = p >> 8, w = p & 255;
  float s = 0.0f;
#pragma unroll
  for (int t = -4; t <= 4; ++t)
    s += GW[t + 4] * y[plane * HW_ + refl256(h + t) * 256 + w];
  tmp[i] = s;
}

__global__ __launch_bounds__(256) void blur_h_acc_k(const float* __restrict__ tmp,
                                                    const float* __restrict__ y,
                                                    float* __restrict__ acc) {
  int i = blockIdx.x * 256 + threadIdx.x;
  int plane = i >> 16;
  int p = i & 65535;
  int h = p >> 8, w = p & 255;
  float s = 0.0f;
#pragma unroll
  for (int t = -4; t <= 4; ++t)
    s += GW[t + 4] * tmp[plane * HW_ + h * 256 + refl256(w + t)];
  float d = s - y[i];
  float v = (0.2f / 393216.0f) * d * d;
  for (int o = 16; o > 0; o >>= 1) v += __shfl_down(v, o, 32);
  if ((threadIdx.x & 31) == 0) atomicAdd(acc, v);
}

__global__ void write_out_k(const float* __restrict__ acc, float* __restrict__ out) {
  out[0] = acc[0];
}

// ---------------------------------------------------------------------------
extern "C" void kernel_launch(void* const* d_in, const int* in_sizes, int n_in,
                              void* d_out, int out_size, void* d_ws, size_t ws_size,
                              hipStream_t stream) {
  (void)in_sizes; (void)n_in; (void)out_size; (void)ws_size;
  const float* y    = (const float*)d_in[0];
  const float* pet  = (const float*)d_in[2];
  const float* mask = (const float*)d_in[3];
  const int*   idx  = (const int*)d_in[5];
  // d_in[1]=mri, d_in[4]=ehat, d_in[6]=epoch: unused by the loss value.

  char* ws = (char*)d_ws;
  float4* yPts = (float4*)(ws + OFF_YPTS);
  float4* pPts = (float4*)(ws + OFF_PPTS);
  float2* potB = (float2*)(ws + OFF_POT);   // 6 arrays of 2*4096 float2
  float2* fP  = potB;                       // f   (y side)
  float2* gP  = potB + 1 * 8192;            // g   (pet side)
  float2* px0 = potB + 2 * 8192;            // sym xx ping (y)
  float2* px1 = potB + 3 * 8192;            // sym xx pong (y)
  float2* py0 = potB + 4 * 8192;            // sym yy ping (pet)
  float2* py1 = potB + 5 * 8192;            // sym yy pong (pet)
  float* acc   = (float*)(ws + OFF_ACC);
  int*   valid = (int*)(ws + OFF_VALID);
  float* tmp   = (float*)(ws + OFF_TMP);

  // zero acc (16 floats) + valid (2 ints) = 18 words; pots seeded by gather
  zero_ws_k<<<1, 32, 0, stream>>>(acc, 18);
  count_valid_k<<<512, 256, 0, stream>>>(mask, valid);
  gather_pts_k<<<32, 256, 0, stream>>>(y, pet, mask, idx, yPts, pPts, potB);

  dim3 pg(32, 2);
  const float wA = 0.8f / 2.0f;      // alpha / B

  // ---- OT(x, y): x = y_s, y = pet_s ----
  for (int it = 0; it < 10; ++it) {
    // g = T(f, C, axis=0): reduce over x rows, output over pet columns
    sink_pass_k<<<pg, 256, 0, stream>>>(yPts, pPts, fP, nullptr, gP, nullptr, nullptr, 0.f, 0);
    // f = T(g, C, axis=1): reduce over pet, output over x
    sink_pass_k<<<pg, 256, 0, stream>>>(pPts, yPts, gP, nullptr, fP, nullptr, nullptr, 0.f, 0);
  }
  // mean(g_fin) + mean(f_fin), weight alpha/B, gated by valid>=32
  sink_pass_k<<<pg, 256, 0, stream>>>(yPts, pPts, fP, nullptr, nullptr, acc, valid, wA, 0);
  sink_pass_k<<<pg, 256, 0, stream>>>(pPts, yPts, gP, nullptr, nullptr, acc, valid, wA, 0);

  // ---- -0.5 * OT(x, x): p <- 0.5*(p + T(p)), 10 iters, then -mean(T(p)) ----
  {
    float2 *cur = px0, *oth = px1;
    for (int it = 0; it < 10; ++it) {
      sink_pass_k<<<pg, 256, 0, stream>>>(yPts, yPts, cur, cur, oth, nullptr, nullptr, 0.f, 1);
      float2* t2 = cur; cur = oth; oth = t2;
    }
    sink_pass_k<<<pg, 256, 0, stream>>>(yPts, yPts, cur, nullptr, nullptr, acc, valid, -wA, 0);
  }
  // ---- -0.5 * OT(y, y) ----
  {
    float2 *cur = py0, *oth = py1;
    for (int it = 0; it < 10; ++it) {
      sink_pass_k<<<pg, 256, 0, stream>>>(pPts, pPts, cur, cur, oth, nullptr, nullptr, 0.f, 1);
      float2* t2 = cur; cur = oth; oth = t2;
    }
    sink_pass_k<<<pg, 256, 0, stream>>>(pPts, pPts, cur, nullptr, nullptr, acc, valid, -wA, 0);
  }

  // ---- beta*L_out + L_intensity ----
  elem_loss_k<<<1536, 256, 0, stream>>>(y, pet, mask, acc);
  // ---- 0.2 * L_smooth ----
  blur_v_k<<<1536, 256, 0, stream>>>(y, tmp);
  blur_h_acc_k<<<1536, 256, 0, stream>>>(tmp, y, acc);

  write_out_k<<<1, 1, 0, stream>>>(acc, (float*)d_out);
}